// MambaExpert_40303973106287
// MI455X (gfx1250) — compile-verified
//
#include <hip/hip_runtime.h>
#include <hip/hip_bf16.h>
#include <math.h>

typedef float v2f __attribute__((ext_vector_type(2)));
typedef float v8f __attribute__((ext_vector_type(8)));
typedef unsigned int v4u __attribute__((ext_vector_type(4)));
typedef int v4i __attribute__((ext_vector_type(4)));
typedef int v8i __attribute__((ext_vector_type(8)));

#define BB      16
#define LL      512
#define CIN     32
#define DMODEL  128
#define DINNER  256
#define DCONV   4
#define DSTATE  16
#define DTRANK  8
#define NXP     40              // DT_RANK + 2*D_STATE
#define COUT    3072            // PRED_LEN * C_IN
#define NROWS   (BB*LL)         // 8192
#define KBIG    (LL*DMODEL)     // 65536
#define KSPLIT  64
#define KCHUNK  (KBIG/KSPLIT)   // 1024
#define NTG     4               // n-tiles (of 16 cols) per wave in big GEMM
#define TILE_COLS (NTG*16)      // 64
#define KB      32              // K rows per TDM tile
#define NBLK    (KCHUNK/KB)     // 32 TDM tiles per wave

// ---------------------------------------------------------------- WMMA helper
__device__ __forceinline__ v8f wmma_f32(v2f a, v2f b, v8f c) {
    // V_WMMA_F32_16X16X4_F32 : D = A(16x4) * B(4x16) + C(16x16), all f32
    return __builtin_amdgcn_wmma_f32_16x16x4_f32(
        /*neg_a=*/false, a, /*neg_b=*/false, b,
        /*c_mod=*/(short)0, c, /*reuse_a=*/false, /*reuse_b=*/false);
}

// ------------------------------------------------- TDM: async 2D tile -> LDS
// D# per CDNA5 ISA ch.8: group0 = {count, lds_addr, global_addr, type=2},
// group1 = {data_size=4B, tensor dims (remaining extent), tile 64x32, stride}.
__device__ __forceinline__ void tdm_load_w_tile(const float* __restrict__ W,
                                                int krow, int col0, void* ldsDst)
{
    unsigned lds_off = (unsigned)(size_t)ldsDst;           // addr[31:0] = LDS offset
    unsigned long long ga =
        (unsigned long long)(size_t)(W + (size_t)krow * COUT + col0);
    unsigned dim0 = (unsigned)(COUT - col0);               // remaining cols
    unsigned dim1 = (unsigned)(KBIG - krow);               // remaining rows
    v4u g0;
    g0.x = 1u;                                             // count=1 (user D#)
    g0.y = lds_off;                                        // lds_addr (bytes)
    g0.z = (unsigned)(ga & 0xffffffffu);                   // global_addr[31:0]
    g0.w = (unsigned)((ga >> 32) & 0x01ffffffu) | (2u << 30);  // addr[56:32]|type=2
    v8i g1;
    g1[0] = (int)(2u << 16);                               // data_size=2 (4B), wg_mask=0
    g1[1] = (int)((dim0 & 0xffffu) << 16);                 // tensor_dim0 lo16
    g1[2] = (int)((dim0 >> 16) | ((dim1 & 0xffffu) << 16));// dim0 hi16 | dim1 lo16
    g1[3] = (int)(((dim1 >> 16) & 0xffffu) | ((unsigned)TILE_COLS << 16)); // dim1 hi | tile_dim0
    g1[4] = KB;                                            // tile_dim1=32, tile_dim2=0
    g1[5] = COUT;                                          // tensor_dim0_stride lo32
    g1[6] = 0;                                             // stride hi16 / dim1_stride
    g1[7] = 0;
    v4i g2 = {0, 0, 0, 0};                                 // 2D tensor: groups 2/3 unused
    v4i g3 = {0, 0, 0, 0};
#if defined(__clang_major__) && (__clang_major__ >= 23)
    v8i g4 = {0, 0, 0, 0, 0, 0, 0, 0};
    __builtin_amdgcn_tensor_load_to_lds(g0, g1, g2, g3, g4, 0);
#else
    __builtin_amdgcn_tensor_load_to_lds(g0, g1, g2, g3, 0);
#endif
}

// ------------------------------------------------- K1: x@W_in -> h@W_inproj
__global__ __launch_bounds__(32) void k_front(
    const float* __restrict__ x, const float* __restrict__ W_in,
    const float* __restrict__ b_in, const float* __restrict__ W_inproj,
    float* __restrict__ u_pre, float* __restrict__ zb)
{
    __shared__ float hS[16][128];
    const int lane = threadIdx.x, l = lane & 15, hi = lane >> 4;
    const int r0 = blockIdx.x * 16;

    // h = x @ W_in + b_in   (16x32 @ 32x128)
    v8f acc[8];
    for (int n = 0; n < 8; ++n) {
        v8f c = {};
        for (int k = 0; k < CIN; k += 4) {
            v2f a = *(const v2f*)(x + (size_t)(r0 + l) * CIN + k + 2 * hi);
            v2f b;
            b.x = W_in[(k + 2*hi    ) * DMODEL + n*16 + l];
            b.y = W_in[(k + 2*hi + 1) * DMODEL + n*16 + l];
            c = wmma_f32(a, b, c);
        }
        acc[n] = c;
    }
    for (int n = 0; n < 8; ++n) {
        float bias = b_in[n*16 + l];
        #pragma unroll
        for (int v = 0; v < 8; ++v)
            hS[v + 8*hi][n*16 + l] = acc[n][v] + bias;
    }
    __syncthreads();

    // xz = h @ W_inproj   (16x128 @ 128x512) ; cols [0,256)->u_pre, rest->z
    for (int n2 = 0; n2 < 32; ++n2) {
        v8f c = {};
        for (int k = 0; k < DMODEL; k += 4) {
            v2f a = *(const v2f*)(&hS[l][k + 2*hi]);
            v2f b;
            b.x = W_inproj[(k + 2*hi    ) * 512 + n2*16 + l];
            b.y = W_inproj[(k + 2*hi + 1) * 512 + n2*16 + l];
            c = wmma_f32(a, b, c);
        }
        float* dst = (n2 < 16) ? (u_pre + n2*16) : (zb + (n2-16)*16);
        #pragma unroll
        for (int v = 0; v < 8; ++v)
            dst[(size_t)(r0 + v + 8*hi) * DINNER + l] = c[v];
    }
}

// ------------------------------------------------- K2: depthwise conv + SiLU
__global__ void k_conv(const float* __restrict__ u_pre,
                       const float* __restrict__ conv_w,
                       const float* __restrict__ conv_b,
                       float* __restrict__ u)
{
    int idx = blockIdx.x * blockDim.x + threadIdx.x;
    if (idx >= NROWS * DINNER) return;
    int d = idx & (DINNER - 1);
    int row = idx >> 8;              // b*L + l
    int l = row & (LL - 1);
    float s = conv_b[d];
    #pragma unroll
    for (int j = 0; j < DCONV; ++j) {
        int li = l - (DCONV - 1) + j;
        float v = (li >= 0) ? u_pre[(size_t)(row - (DCONV - 1) + j) * DINNER + d] : 0.f;
        s = fmaf(conv_w[d * DCONV + j], v, s);
    }
    u[idx] = s / (1.f + __expf(-s));     // SiLU
}

// ------------------------------------------------- K3: xdbc = u @ W_xproj (256->40)
__global__ __launch_bounds__(64) void k_xproj(
    const float* __restrict__ u, const float* __restrict__ W_xproj,
    float* __restrict__ xdbc)
{
    __shared__ float us[DINNER];
    int row = blockIdx.x;
    for (int k = threadIdx.x; k < DINNER; k += 64)
        us[k] = u[(size_t)row * DINNER + k];
    __syncthreads();
    int j = threadIdx.x;
    if (j < NXP) {
        float s = 0.f;
        for (int k = 0; k < DINNER; ++k)
            s = fmaf(us[k], W_xproj[k * NXP + j], s);
        xdbc[(size_t)row * NXP + j] = s;
    }
}

// ------------------------------------------------- K4: dt = softplus(...)
__global__ void k_dt(const float* __restrict__ xdbc, const float* __restrict__ W_dt,
                     const float* __restrict__ b_dt, float* __restrict__ dt)
{
    int idx = blockIdx.x * blockDim.x + threadIdx.x;
    if (idx >= NROWS * DINNER) return;
    int d = idx & (DINNER - 1);
    int row = idx >> 8;
    const float* xr = xdbc + (size_t)row * NXP;
    float s = b_dt[d];
    #pragma unroll
    for (int r = 0; r < DTRANK; ++r)
        s = fmaf(xr[r], W_dt[r * DINNER + d], s);
    dt[idx] = (s > 20.f) ? s : log1pf(__expf(s));
}

// ------------------------------------------------- K5: selective scan (seq over L)
__global__ __launch_bounds__(256) void k_scan(
    const float* __restrict__ dt, const float* __restrict__ u,
    const float* __restrict__ zb, const float* __restrict__ xdbc,
    const float* __restrict__ A_log, const float* __restrict__ Dp,
    float* __restrict__ y)
{
    int b = blockIdx.x, d = threadIdx.x;
    __shared__ float sB[DSTATE], sC[DSTATE];
    float A[DSTATE], st[DSTATE];
    #pragma unroll
    for (int n = 0; n < DSTATE; ++n) {
        A[n]  = -__expf(A_log[d * DSTATE + n]);
        st[n] = 0.f;
    }
    float Dd = Dp[d];
    for (int l = 0; l < LL; ++l) {
        int row = b * LL + l;
        if (d < DSTATE)            sB[d]      = xdbc[(size_t)row * NXP + DTRANK + d];
        else if (d < 2 * DSTATE)   sC[d - 16] = xdbc[(size_t)row * NXP + DTRANK + DSTATE + (d - 16)];
        __syncthreads();
        float dtv = dt[(size_t)row * DINNER + d];
        float uv  = u [(size_t)row * DINNER + d];
        float accv = 0.f;
        #pragma unroll
        for (int n = 0; n < DSTATE; ++n) {
            float dA = __expf(dtv * A[n]);
            st[n] = fmaf(dA, st[n], dtv * sB[n] * uv);
            accv  = fmaf(st[n], sC[n], accv);
        }
        float zv = zb[(size_t)row * DINNER + d];
        float silz = zv / (1.f + __expf(-zv));
        y[(size_t)row * DINNER + d] = (accv + uv * Dd) * silz;
        __syncthreads();
    }
}

// ------------------------------------------------- K6: flat = y @ W_out (256->128)
__global__ __launch_bounds__(32) void k_wout(
    const float* __restrict__ y, const float* __restrict__ W_out,
    float* __restrict__ flat)
{
    const int lane = threadIdx.x, l = lane & 15, hi = lane >> 4;
    const int r0 = blockIdx.x * 16;
    for (int n = 0; n < 8; ++n) {
        v8f c = {};
        for (int k = 0; k < DINNER; k += 4) {
            v2f a = *(const v2f*)(y + (size_t)(r0 + l) * DINNER + k + 2 * hi);
            v2f b;
            b.x = W_out[(k + 2*hi    ) * DMODEL + n*16 + l];
            b.y = W_out[(k + 2*hi + 1) * DMODEL + n*16 + l];
            c = wmma_f32(a, b, c);
        }
        #pragma unroll
        for (int v = 0; v < 8; ++v)
            flat[(size_t)(r0 + v + 8*hi) * DMODEL + n*16 + l] = c[v];
    }
}

// ------------------------------------------------- K7: pred partials, TDM-streamed
// grid (48, 64): x = group of 4 col-tiles, y = K-split chunk. One wave/block.
// W_outproj (805 MB) is DMA'd tile-by-tile into LDS by the Tensor Data Mover
// (double-buffered, TENSORcnt-paced); flat (4 MB) stays L2-resident.
__global__ __launch_bounds__(32) void k_big(
    const float* __restrict__ flat, const float* __restrict__ W,
    float* __restrict__ part)
{
    __shared__ float sW[2][KB * TILE_COLS];      // 2 x 8 KB ping-pong
    const int lane = threadIdx.x, l = lane & 15, hi = lane >> 4;
    const int g  = blockIdx.x;                   // 0..47
    const int ky = blockIdx.y;                   // 0..63
    const int k0 = ky * KCHUNK;
    const int col0 = g * TILE_COLS;
    v8f c[NTG] = {};

    tdm_load_w_tile(W, k0, col0, (void*)&sW[0][0]);
    for (int kb = 0; kb < NBLK; ++kb) {
        const float* buf = &sW[kb & 1][0];
        if (kb + 1 < NBLK) {
            // next buffer was consumed an iteration ago; ensure its ds reads retired
            asm volatile("s_wait_dscnt 0" ::: "memory");
            tdm_load_w_tile(W, k0 + (kb + 1) * KB, col0, (void*)&sW[(kb + 1) & 1][0]);
            __builtin_prefetch(flat + (size_t)l * KBIG + k0 + (kb + 1) * KB, 0, 1);
            __builtin_amdgcn_s_wait_tensorcnt(1);   // current tile landed in LDS
        } else {
            __builtin_amdgcn_s_wait_tensorcnt(0);
        }
        asm volatile("" ::: "memory");
        #pragma unroll
        for (int kk = 0; kk < KB; kk += 4) {
            int kglob = k0 + kb * KB + kk;
            v2f a = *(const v2f*)(flat + (size_t)l * KBIG + kglob + 2 * hi);
            #pragma unroll
            for (int j = 0; j < NTG; ++j) {
                v2f b;
                b.x = buf[(kk + 2*hi    ) * TILE_COLS + j*16 + l];
                b.y = buf[(kk + 2*hi + 1) * TILE_COLS + j*16 + l];
                c[j] = wmma_f32(a, b, c[j]);
            }
        }
    }
    #pragma unroll
    for (int j = 0; j < NTG; ++j)
        #pragma unroll
        for (int v = 0; v < 8; ++v)
            part[(size_t)(ky * 16 + v + 8*hi) * COUT + col0 + j*16 + l] = c[j][v];
}

// ------------------------------------------------- K8: split-K reduce + bias
__global__ void k_reduce(const float* __restrict__ part,
                         const float* __restrict__ bias,
                         float* __restrict__ out)
{
    int idx = blockIdx.x * blockDim.x + threadIdx.x;
    if (idx >= BB * COUT) return;
    int m = idx / COUT, col = idx - m * COUT;
    float s = bias[col];
    for (int ky = 0; ky < KSPLIT; ++ky)
        s += part[(size_t)(ky * 16 + m) * COUT + col];
    out[idx] = s;
}

// ---------------------------------------------------------------- launcher
extern "C" void kernel_launch(void* const* d_in, const int* in_sizes, int n_in,
                              void* d_out, int out_size, void* d_ws, size_t ws_size,
                              hipStream_t stream)
{
    const float* x        = (const float*)d_in[0];
    const float* W_in     = (const float*)d_in[1];
    const float* b_in     = (const float*)d_in[2];
    const float* W_inproj = (const float*)d_in[3];
    const float* conv_w   = (const float*)d_in[4];
    const float* conv_b   = (const float*)d_in[5];
    const float* W_xproj  = (const float*)d_in[6];
    const float* W_dt     = (const float*)d_in[7];
    const float* b_dt     = (const float*)d_in[8];
    const float* A_log    = (const float*)d_in[9];
    const float* Dp       = (const float*)d_in[10];
    const float* W_out    = (const float*)d_in[11];
    const float* W_oproj  = (const float*)d_in[12];
    const float* b_oproj  = (const float*)d_in[13];
    float* out = (float*)d_out;

    // workspace layout (floats)
    float* ws    = (float*)d_ws;
    float* u_pre = ws;                        // 2097152
    float* zb    = u_pre + (size_t)NROWS * DINNER;
    float* u     = zb    + (size_t)NROWS * DINNER;
    float* xdbc  = u     + (size_t)NROWS * DINNER;   // 327680
    float* dt    = xdbc  + (size_t)NROWS * NXP;
    float* y     = dt    + (size_t)NROWS * DINNER;
    float* flat  = y     + (size_t)NROWS * DINNER;   // 1048576
    float* part  = flat  + (size_t)NROWS * DMODEL;   // 3145728

    k_front <<<NROWS / 16, 32, 0, stream>>>(x, W_in, b_in, W_inproj, u_pre, zb);
    k_conv  <<<(NROWS * DINNER) / 256, 256, 0, stream>>>(u_pre, conv_w, conv_b, u);
    k_xproj <<<NROWS, 64, 0, stream>>>(u, W_xproj, xdbc);
    k_dt    <<<(NROWS * DINNER) / 256, 256, 0, stream>>>(xdbc, W_dt, b_dt, dt);
    k_scan  <<<BB, DINNER, 0, stream>>>(dt, u, zb, xdbc, A_log, Dp, y);
    k_wout  <<<NROWS / 16, 32, 0, stream>>>(y, W_out, flat);
    k_big   <<<dim3(COUT / TILE_COLS, KSPLIT), 32, 0, stream>>>(flat, W_oproj, part);
    k_reduce<<<(BB * COUT + 255) / 256, 256, 0, stream>>>(part, b_oproj, out);
}